// LSTMLM_73942156967983
// MI455X (gfx1250) — compile-verified
//
#include <hip/hip_runtime.h>
#include <math.h>

#define VOCAB 50257
#define DIM   128
#define SEQ   2048
#define NGATE (4 * DIM)   // 512

typedef float v2f __attribute__((ext_vector_type(2)));
typedef float v8f __attribute__((ext_vector_type(8)));

__device__ __forceinline__ float sigmoidf_(float x) { return 1.0f / (1.0f + __expf(-x)); }

// ---------------------------------------------------------------------------
// Kernel 1: ifgo_x[t][r] = bias[r] + W_ih[r,:] . E[seq[t],:]   (no recurrence)
// grid = SEQ*2 blocks of 256 threads (each block: one t, 256 of the 512 rows)
// ---------------------------------------------------------------------------
__global__ void __launch_bounds__(256) k_gemmx(const int* __restrict__ seq,
                                               const float* __restrict__ E,
                                               const float* __restrict__ Wih,
                                               const float* __restrict__ bias,
                                               float* __restrict__ ifgo_x) {
    __shared__ __align__(16) float xs[DIM];
    const int t    = blockIdx.x >> 1;
    const int half = blockIdx.x & 1;
    const int tid  = threadIdx.x;
    const int tok  = seq[t];
    if (tid < DIM) xs[tid] = E[tok * DIM + tid];
    __syncthreads();
    const int r = half * 256 + tid;
    const float4* w4 = (const float4*)(Wih + r * DIM);
    const float4* x4 = (const float4*)xs;
    float acc = bias[r];
#pragma unroll
    for (int k = 0; k < DIM / 4; ++k) {
        float4 w = w4[k], x = x4[k];
        acc += w.x * x.x + w.y * x.y + w.z * x.z + w.w * x.w;
    }
    ifgo_x[t * NGATE + r] = acc;
}

// ---------------------------------------------------------------------------
// Kernel 2: the sequential LSTM recurrence (1 workgroup, 512 threads).
// Emits H[t,:] = h_t (state BEFORE update, used for logits), dot[t] = x_t.h_t,
// and the final (h, c) into d_out[1..256].
// ---------------------------------------------------------------------------
__global__ void __launch_bounds__(512) k_lstm(const int* __restrict__ seq,
                                              const float* __restrict__ E,
                                              const float* __restrict__ Whh,
                                              const float* __restrict__ ifgo_x,
                                              const float* __restrict__ c0,
                                              float* __restrict__ H,
                                              float* __restrict__ dotv,
                                              float* __restrict__ out) {
    __shared__ __align__(16) float hs[DIM];
    __shared__ __align__(16) float cs[DIM];
    __shared__ __align__(16) float xs[DIM];
    __shared__ __align__(16) float gv[NGATE];
    const int tid = threadIdx.x;
    if (tid < DIM) { float c = c0[tid]; cs[tid] = c; hs[tid] = tanhf(c); }
    __syncthreads();
    const float4* w4 = (const float4*)(Whh + tid * DIM);   // this thread's gate row
    const float4* h4 = (const float4*)hs;
    for (int t = 0; t < SEQ; ++t) {
        const int tok = seq[t];
        if (tid < DIM) { H[t * DIM + tid] = hs[tid]; xs[tid] = E[tok * DIM + tid]; }
        __syncthreads();
        if (tid < 32) {   // wave 0: dot_t = x . h (pre-update h)
            float p = xs[tid]      * hs[tid]      + xs[tid + 32] * hs[tid + 32]
                    + xs[tid + 64] * hs[tid + 64] + xs[tid + 96] * hs[tid + 96];
            for (int off = 16; off > 0; off >>= 1) p += __shfl_down(p, off, 32);
            if (tid == 0) dotv[t] = p;
        }
        float acc = ifgo_x[t * NGATE + tid];
#pragma unroll
        for (int k = 0; k < DIM / 4; ++k) {
            float4 w = w4[k], h = h4[k];
            acc += w.x * h.x + w.y * h.y + w.z * h.z + w.w * h.w;
        }
        gv[tid] = acc;
        __syncthreads();
        if (tid < DIM) {
            float iv = gv[tid], fv = gv[DIM + tid], gg = gv[2 * DIM + tid], ov = gv[3 * DIM + tid];
            float c = sigmoidf_(fv) * cs[tid] + sigmoidf_(iv) * tanhf(gg);
            cs[tid] = c;
            hs[tid] = sigmoidf_(ov) * tanhf(c);
        }
        __syncthreads();
    }
    if (tid < DIM) { out[1 + tid] = hs[tid]; out[1 + DIM + tid] = cs[tid]; }
}

// ---------------------------------------------------------------------------
// Kernel 3 (WMMA): lossterm[t] = logsumexp_v(E . h_t) - dot[t].
// Each block owns 16 timesteps (one N tile); 16 waves stride over all 3142
// vocab tiles; 32 chained V_WMMA_F32_16X16X4_F32 per tile (K = 128);
// per-lane online logsumexp, combined across waves/halves in LDS.
// ---------------------------------------------------------------------------
__global__ void __launch_bounds__(512) k_loss(const float* __restrict__ E,
                                              const float* __restrict__ H,
                                              const float* __restrict__ dotv,
                                              float* __restrict__ lossterm) {
    __shared__ float sm[512];
    __shared__ float ssum[512];
    const int tid  = threadIdx.x;
    const int wave = tid >> 5;
    const int ln   = tid & 31;
    const int lnN  = ln & 15;
    const int hi   = (ln >> 4) & 1;      // lanes 16-31 hold K/M upper halves
    const int t0   = blockIdx.x * 16;

    // B panel: 16 timesteps x K=128 of H, laid out per the 4x16 f32 B fragment.
    v2f b[32];
    const float* hrow = H + (t0 + lnN) * DIM;
#pragma unroll
    for (int kc = 0; kc < 32; ++kc) {
        const int kb = kc * 4 + hi * 2;
        b[kc].x = hrow[kb];
        b[kc].y = hrow[kb + 1];
    }

    float m = -INFINITY, s = 0.0f;
    const int NTILE = (VOCAB + 15) / 16;          // 3142
    for (int tile = wave; tile < NTILE; tile += 16) {
        const int vbase = tile * 16;
        const int rA    = vbase + lnN;
        const float* arow = E + (rA < VOCAB ? rA : 0) * DIM;   // masked later
        v8f acc = {};
#pragma unroll
        for (int kc = 0; kc < 32; ++kc) {
            const int kb = kc * 4 + hi * 2;
            v2f a;
            a.x = arow[kb];
            a.y = arow[kb + 1];
            acc = __builtin_amdgcn_wmma_f32_16x16x4_f32(
                /*neg_a=*/false, a, /*neg_b=*/false, b[kc],
                /*c_mod=*/(short)0, acc, /*reuse_a=*/false, /*reuse_b=*/false);
        }
        // Online logsumexp over this lane's 8 vocab rows (mask padded rows).
        float lm = -INFINITY;
#pragma unroll
        for (int r = 0; r < 8; ++r) {
            const int row = vbase + (hi ? 8 + r : r);
            if (row < VOCAB) lm = fmaxf(lm, acc[r]);
        }
        if (lm > -INFINITY) {
            const float nm = fmaxf(m, lm);
            float add = 0.0f;
#pragma unroll
            for (int r = 0; r < 8; ++r) {
                const int row = vbase + (hi ? 8 + r : r);
                if (row < VOCAB) add += __expf(acc[r] - nm);
            }
            s = s * __expf(m - nm) + add;
            m = nm;
        }
    }
    sm[tid] = m;
    ssum[tid] = s;
    __syncthreads();

    if (tid < 16) {   // combine 16 waves x 2 lane-halves for timestep t0+tid
        float mm = -INFINITY, as = 0.0f;
        for (int w = 0; w < 16; ++w) {
            for (int hh = 0; hh < 2; ++hh) {
                const int idx = w * 32 + hh * 16 + tid;
                const float mp = sm[idx], sp = ssum[idx];
                const float nm = fmaxf(mm, mp);
                as = as * __expf(mm - nm) + sp * __expf(mp - nm);
                mm = nm;
            }
        }
        const float lse = mm + __logf(as);
        lossterm[t0 + tid] = lse - dotv[t0 + tid];
    }
}

// ---------------------------------------------------------------------------
// Kernel 4: deterministic fixed-order final reduction -> d_out[0].
// ---------------------------------------------------------------------------
__global__ void k_final(const float* __restrict__ lossterm, float* __restrict__ out) {
    const int ln = threadIdx.x;   // 32 threads, one wave
    float s = 0.0f;
    for (int t = ln; t < SEQ; t += 32) s += lossterm[t];
    for (int off = 16; off > 0; off >>= 1) s += __shfl_down(s, off, 32);
    if (ln == 0) out[0] = s;
}

// ---------------------------------------------------------------------------
extern "C" void kernel_launch(void* const* d_in, const int* in_sizes, int n_in,
                              void* d_out, int out_size, void* d_ws, size_t ws_size,
                              hipStream_t stream) {
    (void)in_sizes; (void)n_in; (void)out_size; (void)ws_size;
    const int*   seq  = (const int*)d_in[0];
    const float* E    = (const float*)d_in[1];
    const float* Wih  = (const float*)d_in[2];
    const float* Whh  = (const float*)d_in[3];
    const float* bias = (const float*)d_in[4];
    const float* c0   = (const float*)d_in[5];
    float* out = (float*)d_out;
    float* ws  = (float*)d_ws;

    float* H        = ws;                    // SEQ*DIM        = 262144 floats
    float* ifgo_x   = H + SEQ * DIM;         // SEQ*NGATE      = 1048576 floats
    float* dotv     = ifgo_x + SEQ * NGATE;  // SEQ
    float* lossterm = dotv + SEQ;            // SEQ            (~5.3 MB total)

    k_gemmx<<<SEQ * 2, 256, 0, stream>>>(seq, E, Wih, bias, ifgo_x);
    k_lstm <<<1, 512, 0, stream>>>(seq, E, Whh, ifgo_x, c0, H, dotv, out);
    k_loss <<<SEQ / 16, 512, 0, stream>>>(E, H, dotv, lossterm);
    k_final<<<1, 32, 0, stream>>>(lossterm, out);
}